// STDWetDry_40561671143998
// MI455X (gfx1250) — compile-verified
//
#include <hip/hip_runtime.h>
#include <hip/hip_bf16.h>

// Problem constants from the reference
#define NSTEPS  32
#define TCOLS   16384
#define PADB    15                      // (NSTEPS-1)/2
#define LVALID  (TCOLS - NSTEPS + 1)    // 16353
#define CHUNK_B 4096                    // bytes advanced per DMA chunk (256 lanes * 16B)

// ---- CDNA5 async LDS<->global DMA helpers (gfx1250) ------------------------
// IOFFSET is applied by hardware to BOTH the LDS address and the global
// address, so one base VGPR pair + one LDS-offset VGPR serves all chunks.
template <int OFF>
__device__ __forceinline__ void async_load_off(unsigned lds_base, const float* gbase) {
    asm volatile("global_load_async_to_lds_b128 %0, %1, off offset:%c2 th:TH_LOAD_NT"
                 :: "v"(lds_base), "v"(gbase), "i"(OFF)
                 : "memory");
}

template <int OFF>
__device__ __forceinline__ void async_store_off(const float* gbase, unsigned lds_base) {
    asm volatile("global_store_async_from_lds_b128 %0, %1, off offset:%c2 th:TH_STORE_NT"
                 :: "v"(gbase), "v"(lds_base), "i"(OFF)
                 : "memory");
}

// Compile-time unrolled chains so offsets are true immediates.
template <int K>
struct LoadChain {
    static __device__ __forceinline__ void run(unsigned l, const float* g) {
        LoadChain<K - 1>::run(l, g);
        async_load_off<(K - 1) * CHUNK_B>(l, g);
    }
};
template <> struct LoadChain<0> {
    static __device__ __forceinline__ void run(unsigned, const float*) {}
};

template <int K>
struct StoreChain {
    static __device__ __forceinline__ void run(const float* g, unsigned l) {
        StoreChain<K - 1>::run(g, l);
        async_store_off<(K - 1) * CHUNK_B>(g, l);
    }
};
template <> struct StoreChain<0> {
    static __device__ __forceinline__ void run(const float*, unsigned) {}
};

__device__ __forceinline__ void wait_async0() {
#if defined(__has_builtin) && __has_builtin(__builtin_amdgcn_s_wait_asynccnt)
    __builtin_amdgcn_s_wait_asynccnt(0);
#else
    asm volatile("s_wait_asynccnt 0" ::: "memory");
#endif
}

// LDS byte offset (group-relative): low 32 bits of a generic LDS pointer.
__device__ __forceinline__ unsigned lds_off(const void* p) {
    return (unsigned)(size_t)p;
}

// ---------------------------------------------------------------------------
// One block per row. Stage the 64KB row into LDS via async DMA, compute the
// 32-tap sliding mean/var per thread (64 contiguous columns each) from LDS,
// stage both outputs in LDS, then async-store them back coalesced.
// ---------------------------------------------------------------------------
__global__ __launch_bounds__(256)
void STDWetDry_kernel(const float* __restrict__ x,
                      float* __restrict__ out_hard,
                      float* __restrict__ out_sigma) {
    extern __shared__ float smem[];
    float* xs = smem;               // TCOLS floats : input row
    float* o0 = smem + TCOLS;       // TCOLS floats : hard output
    float* o1 = smem + 2 * TCOLS;   // TCOLS floats : sigma_n_base output

    const int row = blockIdx.x;
    const int t   = threadIdx.x;
    const float* xrow = x + (size_t)row * TCOLS;

    // ---- Stage input row: 16 x b128 async loads per thread, coalesced.
    // Per-thread base = lane's 16B granule; chunks via immediate offsets.
    LoadChain<16>::run(lds_off(xs + t * 4), xrow + t * 4);
    wait_async0();
    __syncthreads();

    // ---- Sliding 32-tap window mean/var over 64 consecutive columns ----
    const float inv_n  = 1.0f / (float)NSTEPS;
    const float inv2th = 1.0f / (2.0f * 1.1f);           // 1/(2*TH)
    const int   c0     = t * 64;

    float s1 = 0.0f, s2 = 0.0f;
    bool  have = false;

#pragma unroll 1
    for (int c = c0; c < c0 + 64; ++c) {
        const int j = c - PADB;                           // window start in sigma space
        float sig = 0.0f;
        if (j >= 0 && j < LVALID) {
            if (!have) {
                s1 = 0.0f; s2 = 0.0f;
#pragma unroll
                for (int kk = 0; kk < NSTEPS; ++kk) {
                    const float v = xs[j + kk];
                    s1 += v;
                    s2 = fmaf(v, v, s2);
                }
                have = true;
            } else {
                const float vn = xs[j + NSTEPS - 1];
                const float vo = xs[j - 1];
                s1 += vn - vo;
                s2 += fmaf(vn, vn, -(vo * vo));
            }
            const float mean = s1 * inv_n;
            const float var  = fmaxf(fmaf(-mean, mean, s2 * inv_n), 0.0f);
            sig = sqrtf(var);
        }
        const float sn   = sig * inv2th;
        const float hard = fminf(fmaxf(rintf(sn), 0.0f), 1.0f);  // round-half-even like jnp.round
        o0[c] = hard;   // out = sn + stopgrad(hard - sn) == hard (forward value)
        o1[c] = sig;    // sigma_n_base (padded sigma)
    }
    __syncthreads();

    // ---- Stream both outputs out with coalesced NT async stores from LDS ----
    const float* oh = out_hard  + (size_t)row * TCOLS;
    const float* os = out_sigma + (size_t)row * TCOLS;
    StoreChain<16>::run(oh + t * 4, lds_off(o0 + t * 4));
    StoreChain<16>::run(os + t * 4, lds_off(o1 + t * 4));
    wait_async0();   // (S_ENDPGM also implies wait-idle; explicit for clarity)
}

extern "C" void kernel_launch(void* const* d_in, const int* in_sizes, int n_in,
                              void* d_out, int out_size, void* d_ws, size_t ws_size,
                              hipStream_t stream) {
    const float* x = (const float*)d_in[0];
    float* out = (float*)d_out;

    const int B = in_sizes[0] / TCOLS;                    // 2048
    float* out_hard  = out;                               // first output, B*T floats
    float* out_sigma = out + (size_t)B * TCOLS;           // second output, B*T floats

    const size_t shmem = (size_t)3 * TCOLS * sizeof(float);   // 192 KB/WG (<=320KB on CDNA5)
    dim3 grid(B), block(256);
    hipLaunchKernelGGL(STDWetDry_kernel, grid, block, shmem, stream,
                       x, out_hard, out_sigma);
}